// ChamferDistance_91079076479382
// MI455X (gfx1250) — compile-verified
//
#include <hip/hip_runtime.h>

typedef __attribute__((ext_vector_type(2))) float v2f;
typedef __attribute__((ext_vector_type(8))) float v8f;
typedef __attribute__((__vector_size__(16))) int v4i_vs;   // int4 pointee for async-LDS builtin
typedef __attribute__((address_space(1))) v4i_vs* gv4i_p;  // global int4*
typedef __attribute__((address_space(3))) v4i_vs* lv4i_p;  // LDS int4*

#define NPTS 8192
#define MPTS 8192
#define ROWS_PER_BLOCK 128    // 8 waves * 16 rows
#define BLOCK_THREADS 256
#define BIGF 3.0e38f

#define CHUNK_PTS 1024                    // xyz2 points staged per chunk
#define NCHUNKS   (MPTS / CHUNK_PTS)      // 8
#define STAGE_FLOATS (CHUNK_PTS * 3)      // 3072 floats = 12KB per buffer

// ---- gfx1250 async global->LDS staging (ASYNCcnt-tracked), with fallback ----
#if defined(__has_builtin)
#  if __has_builtin(__builtin_amdgcn_global_load_async_to_lds_b128)
#    define USE_ASYNC_STAGE 1
#  endif
#endif
#ifndef USE_ASYNC_STAGE
#  define USE_ASYNC_STAGE 0
#endif

__device__ __forceinline__ void wait_async_zero() {
#if defined(__has_builtin) && __has_builtin(__builtin_amdgcn_s_wait_asynccnt)
    __builtin_amdgcn_s_wait_asynccnt(0);
#else
    asm volatile("s_wait_asynccnt 0x0" ::: "memory");
#endif
}

__global__ void chamfer_init_kernel(float* __restrict__ out, int n) {
    int i = blockIdx.x * blockDim.x + threadIdx.x;
    if (i < n) out[i] = BIGF;
}

__global__ __launch_bounds__(BLOCK_THREADS)
void ChamferDistance_91079076479382_kernel(const float* __restrict__ xyz1,
                                           const float* __restrict__ xyz2,
                                           float* __restrict__ dist1,
                                           float* __restrict__ dist2) {
    __shared__ float colmin[MPTS];                 // 32KB per-block column mins
#if USE_ASYNC_STAGE
    __shared__ float stage[2][STAGE_FLOATS];       // 24KB double-buffered xyz2 chunks
#else
    __shared__ float stage[1][STAGE_FLOATS];
#endif

    const int b    = blockIdx.y;
    const int tid  = threadIdx.x;
    const int wave = tid >> 5;              // wave32
    const int lane = tid & 31;
    const int lr   = lane & 15;             // position within 16-lane half
    const bool hi  = lane >= 16;            // upper half: K=2..3 / rows 8..15

    for (int i = tid; i < MPTS; i += BLOCK_THREADS) colmin[i] = BIGF;

    const float* x1 = xyz1 + (size_t)b * NPTS * 3;
    const float* x2 = xyz2 + (size_t)b * MPTS * 3;

    const int tile_n = blockIdx.x * ROWS_PER_BLOCK + wave * 16;

    // ---- A tile (16x4 f32, K=3 padded), pre-scaled by -2 ----
    // lanes 0-15: VGPR0=K0, VGPR1=K1 ; lanes 16-31: VGPR0=K2, VGPR1=K3(=0)
    const int arow = tile_n + lr;
    const float ax = x1[arow * 3 + 0];
    const float ay = x1[arow * 3 + 1];
    const float az = x1[arow * 3 + 2];
    v2f a;
    a.x = hi ? az : ax;
    a.y = hi ? 0.0f : ay;
    a *= -2.0f;    // fold the -2 of (aa + bb - 2ab) into A once

    // per-lane aa for the 8 rows this half's C VGPRs cover
    const int rbase = tile_n + (hi ? 8 : 0);
    float aa[8];
#pragma unroll
    for (int r = 0; r < 8; ++r) {
        const float x = x1[(rbase + r) * 3 + 0];
        const float y = x1[(rbase + r) * 3 + 1];
        const float z = x1[(rbase + r) * 3 + 2];
        aa[r] = x * x + y * y + z * z;
    }

    float rowmin[8];
#pragma unroll
    for (int r = 0; r < 8; ++r) rowmin[r] = BIGF;

#if USE_ASYNC_STAGE
    // issue async prefetch of chunk 0 into buffer 0
    {
        const float* src = x2;
        float* dst = &stage[0][0];
#pragma unroll
        for (int i = 0; i < STAGE_FLOATS / 4 / BLOCK_THREADS; ++i) {  // 3 x b128
            const int f4 = tid + i * BLOCK_THREADS;
            __builtin_amdgcn_global_load_async_to_lds_b128(
                (gv4i_p)(src + f4 * 4), (lv4i_p)(dst + f4 * 4), 0, 0);
        }
    }
#endif

    int buf = 0;
    for (int chunk = 0; chunk < NCHUNKS; ++chunk) {
#if USE_ASYNC_STAGE
        wait_async_zero();        // my async copies for stage[buf] complete
        __syncthreads();          // everyone's copies visible; old buffer free
        if (chunk + 1 < NCHUNKS) {
            const float* src = x2 + (size_t)(chunk + 1) * STAGE_FLOATS;
            float* dst = &stage[buf ^ 1][0];
#pragma unroll
            for (int i = 0; i < STAGE_FLOATS / 4 / BLOCK_THREADS; ++i) {
                const int f4 = tid + i * BLOCK_THREADS;
                __builtin_amdgcn_global_load_async_to_lds_b128(
                    (gv4i_p)(src + f4 * 4), (lv4i_p)(dst + f4 * 4), 0, 0);
            }
        }
#else
        __syncthreads();          // previous chunk's readers done
        {
            const float4* src = (const float4*)(x2 + (size_t)chunk * STAGE_FLOATS);
            float4* dst = (float4*)&stage[0][0];
#pragma unroll
            for (int i = 0; i < STAGE_FLOATS / 4 / BLOCK_THREADS; ++i)
                dst[tid + i * BLOCK_THREADS] = src[tid + i * BLOCK_THREADS];
        }
        __syncthreads();
#endif

        const float* sb = &stage[buf][0];
        const int mbase = chunk * CHUNK_PTS;

        for (int t = 0; t < CHUNK_PTS; t += 16) {
            const int p  = t + lr;              // point within chunk (LDS)
            const float bx = sb[p * 3 + 0];
            const float by = sb[p * 3 + 1];
            const float bz = sb[p * 3 + 2];
            const float bb = fmaf(bx, bx, fmaf(by, by, bz * bz));

            v2f bv;
            bv.x = hi ? bz : bx;
            bv.y = hi ? 0.0f : by;

            v8f c = {};
            // c = (-2*A) x B : 16x16 tile of -2 * <x1_n, x2_m>
            c = __builtin_amdgcn_wmma_f32_16x16x4_f32(
                    /*neg_a=*/false, a, /*neg_b=*/false, bv,
                    /*c_mod=*/(short)0, c, /*reuse_a=*/false, /*reuse_b=*/false);

            float colm = BIGF;
#pragma unroll
            for (int r = 0; r < 8; ++r) {
                const float d = (aa[r] + bb) + c[r];    // aa + bb - 2ab
                rowmin[r] = fminf(rowmin[r], d);
                colm = fminf(colm, d);
            }
            // all 32 lanes: lanes L and L+16 min into the same column entry
            // (each carries the min over its 8-row half) -> single ds_min_num_f32
            atomicMin(&colmin[mbase + p], colm);
        }

#if USE_ASYNC_STAGE
        buf ^= 1;
#endif
    }

    // ---- dist1: reduce row mins across the 16 lanes of each half ----
#pragma unroll
    for (int r = 0; r < 8; ++r) {
        float v = rowmin[r];
        v = fminf(v, __shfl_xor(v, 1, 32));
        v = fminf(v, __shfl_xor(v, 2, 32));
        v = fminf(v, __shfl_xor(v, 4, 32));
        v = fminf(v, __shfl_xor(v, 8, 32));
        if (lr == 0)   // lane 0 -> rows 0..7, lane 16 -> rows 8..15
            dist1[(size_t)b * NPTS + rbase + r] = v;
    }

    // ---- dist2: flush block-local column mins with global float atomic min ----
    __syncthreads();
    float* d2 = dist2 + (size_t)b * MPTS;
    for (int i = tid; i < MPTS; i += BLOCK_THREADS)
        atomicMin(&d2[i], colmin[i]);
}

extern "C" void kernel_launch(void* const* d_in, const int* in_sizes, int n_in,
                              void* d_out, int out_size, void* d_ws, size_t ws_size,
                              hipStream_t stream) {
    const float* xyz1 = (const float*)d_in[0];
    const float* xyz2 = (const float*)d_in[1];
    float* out   = (float*)d_out;
    const int B  = in_sizes[0] / (NPTS * 3);       // = 4
    float* dist1 = out;                            // [B, N]
    float* dist2 = out + (size_t)B * NPTS;         // [B, M]

    const int total = B * NPTS * 2;
    chamfer_init_kernel<<<(total + 255) / 256, 256, 0, stream>>>(out, total);

    dim3 grid(NPTS / ROWS_PER_BLOCK, B);           // (64, 4)
    ChamferDistance_91079076479382_kernel<<<grid, BLOCK_THREADS, 0, stream>>>(
        xyz1, xyz2, dist1, dist2);
}